// RefinedLayer_60773787238719
// MI455X (gfx1250) — compile-verified
//
#include <hip/hip_runtime.h>
#include <hip/hip_bf16.h>

#define NN 50000
#define EE 800000
#define HIDD 128
#define GAMMA_C 1.0f
#define DELTA_C 0.5f
#define LN_EPS_C 1e-5f

typedef float v2f __attribute__((ext_vector_type(2)));
typedef float v8f __attribute__((ext_vector_type(8)));

// ---------- helpers ----------
__device__ __forceinline__ unsigned fkey(float f) {
    unsigned b = __float_as_uint(f);
    return (b & 0x80000000u) ? ~b : (b | 0x80000000u);
}
__device__ __forceinline__ float fdec(unsigned k) {
    return (k & 0x80000000u) ? __uint_as_float(k & 0x7FFFFFFFu)
                             : __uint_as_float(~k);
}
__device__ __forceinline__ float wred_sum(float v) {
#pragma unroll
    for (int off = 16; off > 0; off >>= 1) v += __shfl_xor(v, off, 32);
    return v;
}

// ---------- zero fill ----------
__global__ void zero_f32(float* __restrict__ p, size_t n) {
    size_t i = (size_t)blockIdx.x * blockDim.x + threadIdx.x;
    if (i < n) p[i] = 0.0f;
}

// ---------- WMMA f32 GEMM: D[M x 128] = A[M x 128] @ B[128 x 128] (+bias) ----------
// grid.x = M/16 row slabs, 256 threads = 8 waves, wave w -> 16-col tile w.
__global__ __launch_bounds__(256) void gemm_wmma_f32(
    const float* __restrict__ A, const float* __restrict__ B,
    const float* __restrict__ bias, float* __restrict__ D) {
    const int lane = threadIdx.x & 31;
    const int wave = threadIdx.x >> 5;          // 0..7
    const int row0 = blockIdx.x * 16;
    const int colBase = wave * 16;
    const int r = lane & 15;
    const int khalf = (lane >> 4) << 1;         // 0 or 2

    v8f acc = {};
    const float* Arow = A + (size_t)(row0 + r) * HIDD;
#pragma unroll 4
    for (int k = 0; k < HIDD; k += 4) {
        v2f a, b;
        // A 16x4 frag: lanes 0-15 rows, VGPR0/1 = K pair (khalf selects K 0/1 vs 2/3)
        a.x = Arow[k + khalf];
        a.y = Arow[k + khalf + 1];
        // B 4x16 frag (row-major B[k][n]), mirrored striping
        const int kk = k + khalf;
        b.x = B[(size_t)kk * HIDD + colBase + r];
        b.y = B[(size_t)(kk + 1) * HIDD + colBase + r];
        acc = __builtin_amdgcn_wmma_f32_16x16x4_f32(
            false, a, false, b, (short)0, acc, false, false);
    }
    // D 16x16 f32 tile: VGPR i -> M = i + 8*(lane>=16), N = lane%16
    const int col = colBase + r;
    const int rowOff = (lane >> 4) << 3;        // 0 or 8
    const float bv = bias ? bias[col] : 0.0f;
#pragma unroll
    for (int i = 0; i < 8; ++i) {
        const int row = row0 + rowOff + i;
        D[(size_t)row * HIDD + col] = acc[i] + bv;
    }
}

// ---------- psi[n] = H[n] . f_def_w + f_def_b  (warp per node) ----------
__global__ void psi_kernel(const float* __restrict__ H,
                           const float* __restrict__ fw,
                           const float* __restrict__ fb,
                           float* __restrict__ psi) {
    const int warp = (int)(((size_t)blockIdx.x * blockDim.x + threadIdx.x) >> 5);
    const int lane = threadIdx.x & 31;
    if (warp >= NN) return;
    const float4 hv = ((const float4*)(H + (size_t)warp * HIDD))[lane];
    const float4 wv = ((const float4*)fw)[lane];
    float p = hv.x * wv.x + hv.y * wv.y + hv.z * wv.z + hv.w * wv.w;
    p = wred_sum(p);
    if (lane == 0) psi[warp] = p + fb[0];
}

// ---------- per-edge scores + segment max (warp per edge) ----------
__global__ void edge_score_kernel(const int* __restrict__ ei,
                                  const float* __restrict__ H,
                                  const float* __restrict__ HWatt,
                                  const float* __restrict__ HP,
                                  const float* __restrict__ HPp,
                                  float* __restrict__ scoreA,
                                  float* __restrict__ scoreP,
                                  unsigned* __restrict__ maxA,
                                  unsigned* __restrict__ maxP) {
    const int e = (int)(((size_t)blockIdx.x * blockDim.x + threadIdx.x) >> 5);
    const int lane = threadIdx.x & 31;
    if (e >= EE) return;
    const int src = ei[e];
    const int tgt = ei[EE + e];

    const float4 ht = ((const float4*)(H + (size_t)tgt * HIDD))[lane];
    const float4 wa = ((const float4*)(HWatt + (size_t)src * HIDD))[lane];
    float sA = ht.x * wa.x + ht.y * wa.y + ht.z * wa.z + ht.w * wa.w;
    // structural dot: h[src][0:6] . h[tgt][0:6]
    if (lane == 0) {
        const float4 hs = ((const float4*)(H + (size_t)src * HIDD))[0];
        sA += hs.x * ht.x + hs.y * ht.y + hs.z * ht.z + hs.w * ht.w;
    } else if (lane == 1) {
        const float4 hs = ((const float4*)(H + (size_t)src * HIDD))[1];
        sA += hs.x * ht.x + hs.y * ht.y;   // features 4,5 only
    }
    const float4 p1 = ((const float4*)(HP + (size_t)src * HIDD))[lane];
    const float4 p2 = ((const float4*)(HPp + (size_t)tgt * HIDD))[lane];
    float sP = p1.x * p2.x + p1.y * p2.y + p1.z * p2.z + p1.w * p2.w;

    sA = wred_sum(sA);
    sP = wred_sum(sP);
    if (lane == 0) {
        scoreA[e] = sA;
        scoreP[e] = sP;
        atomicMax(maxA + tgt, fkey(sA));
        atomicMax(maxP + src, fkey(sP));
    }
}

// ---------- exp + segment sums (thread per edge) ----------
__global__ void edge_exp_kernel(const int* __restrict__ ei,
                                const float* __restrict__ psi,
                                const unsigned* __restrict__ maxA,
                                const unsigned* __restrict__ maxP,
                                float* __restrict__ scoreA,   // -> eA
                                float* __restrict__ scoreP,   // -> eP
                                float* __restrict__ sumA,
                                float* __restrict__ sumP,
                                float* __restrict__ numer) {
    const int e = (int)((size_t)blockIdx.x * blockDim.x + threadIdx.x);
    if (e >= EE) return;
    const int src = ei[e];
    const int tgt = ei[EE + e];
    const float eA = __expf(scoreA[e] - fdec(maxA[tgt]));
    const float eP = __expf(scoreP[e] - fdec(maxP[src]));
    scoreA[e] = eA;
    scoreP[e] = eP;
    atomicAdd(sumA + tgt, eA);
    atomicAdd(sumP + src, eP);
    atomicAdd(numer + src, eP * __expf(-psi[tgt]));
}

// ---------- rho per node ----------
__global__ void rho_kernel(const float* __restrict__ sumP,
                           const float* __restrict__ numer,
                           float* __restrict__ rho) {
    const int n = (int)((size_t)blockIdx.x * blockDim.x + threadIdx.x);
    if (n >= NN) return;
    const float sp = sumP[n];
    const float st = (sp > 0.0f) ? (numer[n] / sp) : 0.0f;
    const float d = -__logf(st + 1e-8f);
    const float x = GAMMA_C * (d - DELTA_C);
    rho[n] = 1.0f / (1.0f + __expf(-x));
}

// ---------- message scatter: m_att[tgt] += alpha*(1-rho[src])*Hphi[src] ----------
__global__ void edge_msg_kernel(const int* __restrict__ ei,
                                const float* __restrict__ eA,
                                const float* __restrict__ sumA,
                                const float* __restrict__ rho,
                                const float* __restrict__ Hphi,
                                float* __restrict__ m_att) {
    const int e = (int)(((size_t)blockIdx.x * blockDim.x + threadIdx.x) >> 5);
    const int lane = threadIdx.x & 31;
    if (e >= EE) return;
    const int src = ei[e];
    const int tgt = ei[EE + e];
    const float c = (eA[e] / sumA[tgt]) * (1.0f - rho[src]);
    const float4 ph = ((const float4*)(Hphi + (size_t)src * HIDD))[lane];
    float* dst = m_att + (size_t)tgt * HIDD + lane * 4;
    atomicAdd(dst + 0, c * ph.x);
    atomicAdd(dst + 1, c * ph.y);
    atomicAdd(dst + 2, c * ph.z);
    atomicAdd(dst + 3, c * ph.w);
}

// ---------- fused relu + residual + LayerNorm (warp per node) ----------
__global__ void final_kernel(const float* __restrict__ H,
                             const float* __restrict__ Hpre,   // H@W_self + b
                             const float* __restrict__ mA,     // m_att@W_A + b
                             const float* __restrict__ Wstr,   // [6 x 128]
                             const float* __restrict__ Wstr_b,
                             const float* __restrict__ ln_g,
                             const float* __restrict__ ln_b,
                             float* __restrict__ out) {
    const int n = (int)(((size_t)blockIdx.x * blockDim.x + threadIdx.x) >> 5);
    const int lane = threadIdx.x & 31;
    if (n >= NN) return;
    const int c0 = lane * 4;

    const float4 pre = ((const float4*)(Hpre + (size_t)n * HIDD))[lane];
    const float4 ma  = ((const float4*)(mA + (size_t)n * HIDD))[lane];
    const float4 h4  = ((const float4*)(H + (size_t)n * HIDD))[lane];
    const float4 sb  = ((const float4*)Wstr_b)[lane];

    float4 v;
    v.x = pre.x + ma.x + sb.x;
    v.y = pre.y + ma.y + sb.y;
    v.z = pre.z + ma.z + sb.z;
    v.w = pre.w + ma.w + sb.w;
#pragma unroll
    for (int j = 0; j < 6; ++j) {
        const float hj = H[(size_t)n * HIDD + j];           // warp-uniform
        const float4 wj = ((const float4*)(Wstr + (size_t)j * HIDD))[lane];
        v.x += hj * wj.x; v.y += hj * wj.y; v.z += hj * wj.z; v.w += hj * wj.w;
    }
    v.x = fmaxf(v.x, 0.0f) + h4.x;
    v.y = fmaxf(v.y, 0.0f) + h4.y;
    v.z = fmaxf(v.z, 0.0f) + h4.z;
    v.w = fmaxf(v.w, 0.0f) + h4.w;

    const float mu = wred_sum(v.x + v.y + v.z + v.w) * (1.0f / HIDD);
    float4 dvec;
    dvec.x = v.x - mu; dvec.y = v.y - mu; dvec.z = v.z - mu; dvec.w = v.w - mu;
    const float var = wred_sum(dvec.x * dvec.x + dvec.y * dvec.y +
                               dvec.z * dvec.z + dvec.w * dvec.w) * (1.0f / HIDD);
    const float rs = rsqrtf(var + LN_EPS_C);

    const float4 g = ((const float4*)ln_g)[lane];
    const float4 b = ((const float4*)ln_b)[lane];
    float4 o;
    o.x = dvec.x * rs * g.x + b.x;
    o.y = dvec.y * rs * g.y + b.y;
    o.z = dvec.z * rs * g.z + b.z;
    o.w = dvec.w * rs * g.w + b.w;
    ((float4*)(out + (size_t)n * HIDD))[lane] = o;
}

extern "C" void kernel_launch(void* const* d_in, const int* in_sizes, int n_in,
                              void* d_out, int out_size, void* d_ws, size_t ws_size,
                              hipStream_t stream) {
    const int*   ei       = (const int*)d_in[0];
    const float* H        = (const float*)d_in[1];
    const float* W_att    = (const float*)d_in[2];
    const float* phi_w    = (const float*)d_in[3];
    const float* phi_b    = (const float*)d_in[4];
    const float* W_p      = (const float*)d_in[5];
    const float* W_pp     = (const float*)d_in[6];
    const float* fdef_w   = (const float*)d_in[7];
    const float* fdef_b   = (const float*)d_in[8];
    const float* W_self_w = (const float*)d_in[9];
    const float* W_self_b = (const float*)d_in[10];
    const float* W_A_w    = (const float*)d_in[11];
    const float* W_A_b    = (const float*)d_in[12];
    const float* W_str_w  = (const float*)d_in[13];
    const float* W_str_b  = (const float*)d_in[14];
    const float* ln_g     = (const float*)d_in[15];
    const float* ln_b     = (const float*)d_in[16];
    float* out = (float*)d_out;

    char* ws = (char*)d_ws;
    const size_t NB = (size_t)NN * HIDD * sizeof(float);   // 25.6 MB
    float* HWatt = (float*)(ws + 0 * NB);                  // later reused as mA
    float* HP    = (float*)(ws + 1 * NB);                  // later reused as m_att
    float* HPp   = (float*)(ws + 2 * NB);
    float* Hphi  = (float*)(ws + 3 * NB);
    float* Hpre  = (float*)(ws + 4 * NB);
    char* nb = ws + 5 * NB;
    float*    psi   = (float*)(nb + 0 * (size_t)NN * 4);
    unsigned* maxA  = (unsigned*)(nb + 1 * (size_t)NN * 4);
    unsigned* maxP  = (unsigned*)(nb + 2 * (size_t)NN * 4);
    float*    sumA  = (float*)(nb + 3 * (size_t)NN * 4);
    float*    sumP  = (float*)(nb + 4 * (size_t)NN * 4);
    float*    numer = (float*)(nb + 5 * (size_t)NN * 4);
    float*    rho   = (float*)(nb + 6 * (size_t)NN * 4);
    float*    scoreA = (float*)(nb + 7 * (size_t)NN * 4);
    float*    scoreP = scoreA + EE;

    // zero maxA/maxP/sumA/sumP/numer (contiguous 5*N words; 0u == fkey floor)
    {
        size_t zn = 5 * (size_t)NN;
        zero_f32<<<(unsigned)((zn + 255) / 256), 256, 0, stream>>>(
            (float*)(nb + (size_t)NN * 4), zn);
    }

    // node GEMMs via f32 WMMA
    const dim3 ggrid(NN / 16), gblk(256);
    gemm_wmma_f32<<<ggrid, gblk, 0, stream>>>(H, W_att,    nullptr,  HWatt);
    gemm_wmma_f32<<<ggrid, gblk, 0, stream>>>(H, W_p,      nullptr,  HP);
    gemm_wmma_f32<<<ggrid, gblk, 0, stream>>>(H, W_pp,     nullptr,  HPp);
    gemm_wmma_f32<<<ggrid, gblk, 0, stream>>>(H, phi_w,    phi_b,    Hphi);
    gemm_wmma_f32<<<ggrid, gblk, 0, stream>>>(H, W_self_w, W_self_b, Hpre);

    psi_kernel<<<(NN * 32 + 255) / 256, 256, 0, stream>>>(H, fdef_w, fdef_b, psi);

    edge_score_kernel<<<(EE * 32 + 255) / 256, 256, 0, stream>>>(
        ei, H, HWatt, HP, HPp, scoreA, scoreP, maxA, maxP);

    // m_att aliases HP (HP dead after edge_score); zero it now
    {
        size_t zn = (size_t)NN * HIDD;
        zero_f32<<<(unsigned)((zn + 255) / 256), 256, 0, stream>>>(HP, zn);
    }

    edge_exp_kernel<<<(EE + 255) / 256, 256, 0, stream>>>(
        ei, psi, maxA, maxP, scoreA, scoreP, sumA, sumP, numer);

    rho_kernel<<<(NN + 255) / 256, 256, 0, stream>>>(sumP, numer, rho);

    edge_msg_kernel<<<(EE * 32 + 255) / 256, 256, 0, stream>>>(
        ei, scoreA, sumA, rho, Hphi, /*m_att=*/HP);

    // mA aliases HWatt (dead after edge_score); mA = m_att @ W_A + b
    gemm_wmma_f32<<<ggrid, gblk, 0, stream>>>(/*A=*/HP, W_A_w, W_A_b, /*D=*/HWatt);

    final_kernel<<<(NN * 32 + 255) / 256, 256, 0, stream>>>(
        H, Hpre, /*mA=*/HWatt, W_str_w, W_str_b, ln_g, ln_b, out);
}